// DenseDepthLoss_10943576670673
// MI455X (gfx1250) — compile-verified
//
#include <hip/hip_runtime.h>
#include <math.h>

// ---------------- CDNA5 WMMA types ----------------
typedef __attribute__((ext_vector_type(2))) float v2f;
typedef __attribute__((ext_vector_type(8))) float v8f;

#define BATCH   64
#define IMG_H   480
#define IMG_W   640
#define OUT_H   470
#define OUT_W   630
#define TILE_H  32
#define TILE_W  64
#define TILES_X 10          // ceil(630/64); tiles also exactly partition IMG_W (10*64=640)
#define TILES_Y 15          // ceil(470/32); tiles also exactly partition IMG_H (15*32=480)
#define IN_H    48          // 32 + 16 rows (second h-conv WMMA tile reach)
#define IN_W    76          // 64 + 12 cols (chunk padding to K=28)

#define C1_SSIM 98.01f      // (0.01*990)^2
#define C2_SSIM 882.09f     // (0.03*990)^2
#define C3_SSIM 441.045f

// unnormalized gaussian pdf samples, sigma=1.5, z=-5..5 (matches reference)
__device__ __constant__ float c_g11[11] = {
    1.0282213e-03f, 7.5973736e-03f, 3.5993993e-02f, 1.0934006e-01f,
    2.1296533e-01f, 2.6596152e-01f, 2.1296533e-01f, 1.0934006e-01f,
    3.5993993e-02f, 7.5973736e-03f, 1.0282213e-03f };

__device__ __forceinline__ float gband(int j, int n) {
    int d = j - n;
    return (d >= 0 && d <= 10) ? c_g11[d] : 0.0f;
}

// ---------------- kernel 0: zero the accumulators ----------------
__global__ void dd_init_kernel(float* ws) {
    if (threadIdx.x < 3) ws[threadIdx.x] = 0.0f;
}

// ---------------- kernel 1: fused L1 + gradient + SSIM ----------------
// Per block: 32x64 region, 8 waves.
//  - cooperative load of 48x76 input tile (borders clamped) into LDS
//  - L1 + gradient loss over the block's exact 32x64 ownership region (reuses LDS)
//  - SSIM: separable 11-tap gaussian as WMMA band-matmul, each wave owns a 16x16 tile:
//      h-conv: tmp(16x16) = A_data(16x4) x G_band(4x16)   (7 K-chunks, f32 16x16x4)
//      v-conv: out(16x16) = G_band(16x4) x B_tmp(4x16)    (same gaussian fragment)
__global__ __launch_bounds__(256) void dd_fused_kernel(const float* __restrict__ pred,
                                                       const float* __restrict__ targ,
                                                       float* __restrict__ ws) {
    __shared__ float sP[IN_H][IN_W];
    __shared__ float sT[IN_H][IN_W];
    __shared__ float sTmp[8][32][16];   // per-wave h-conv scratch (one channel at a time)
    __shared__ float sL1, sGR, sSS;

    const int b    = blockIdx.x / (TILES_X * TILES_Y);
    const int tile = blockIdx.x % (TILES_X * TILES_Y);
    const int oy0  = (tile / TILES_X) * TILE_H;
    const int ox0  = (tile % TILES_X) * TILE_W;

    const float* pb = pred + (size_t)b * (IMG_H * IMG_W);
    const float* tb = targ + (size_t)b * (IMG_H * IMG_W);

    if (threadIdx.x == 0) { sL1 = 0.0f; sGR = 0.0f; sSS = 0.0f; }

    // cooperative tile load, borders clamped (keeps padded WMMA K-lanes finite)
    for (int i = threadIdx.x; i < IN_H * IN_W; i += 256) {
        const int i2 = i + 512;                      // prefetch two strides ahead
        if (i2 < IN_H * IN_W) {
            const int r2  = i2 / IN_W;
            const int c2  = i2 - r2 * IN_W;
            const size_t pidx = (size_t)min(oy0 + r2, IMG_H - 1) * IMG_W
                              + min(ox0 + c2, IMG_W - 1);
            __builtin_prefetch(&pb[pidx], 0, 1);     // global_prefetch_b8
            __builtin_prefetch(&tb[pidx], 0, 1);
        }
        const int r  = i / IN_W;
        const int c  = i - r * IN_W;
        const int gy = min(oy0 + r, IMG_H - 1);
        const int gx = min(ox0 + c, IMG_W - 1);
        const size_t idx = (size_t)gy * IMG_W + gx;
        sP[r][c] = pb[idx];
        sT[r][c] = tb[idx];
    }
    __syncthreads();

    // ---- L1 + gradient over this block's exact 32x64 ownership region ----
    // Tiles partition the image, so every pixel is counted exactly once.
    {
        float l1 = 0.0f, gr = 0.0f;
        for (int i = threadIdx.x; i < TILE_H * TILE_W; i += 256) {
            const int r  = i >> 6;          // 0..31  (unclamped in LDS: r < 32 <= 47)
            const int c  = i & 63;          // 0..63  (unclamped in LDS: c < 64 <= 75)
            const int gy = oy0 + r;
            const int gx = ox0 + c;
            const float p = sP[r][c], t = sT[r][c];
            l1 += fabsf(p - t);
            // zero-padded cross-correlation neighbors (XLA conv semantics)
            const float pl = (gx == 0) ? 0.f : (c > 0 ? sP[r][c - 1] : pb[(size_t)gy * IMG_W + gx - 1]);
            const float tl = (gx == 0) ? 0.f : (c > 0 ? sT[r][c - 1] : tb[(size_t)gy * IMG_W + gx - 1]);
            const float pu = (gy == 0) ? 0.f : (r > 0 ? sP[r - 1][c] : pb[(size_t)(gy - 1) * IMG_W + gx]);
            const float tu = (gy == 0) ? 0.f : (r > 0 ? sT[r - 1][c] : tb[(size_t)(gy - 1) * IMG_W + gx]);
            const float pr = (gx == IMG_W - 1) ? 0.f : sP[r][c + 1];
            const float tr = (gx == IMG_W - 1) ? 0.f : sT[r][c + 1];
            const float pd = (gy == IMG_H - 1) ? 0.f : sP[r + 1][c];
            const float td = (gy == IMG_H - 1) ? 0.f : sT[r + 1][c];
            gr += fabsf((pr - pl) - (tr - tl)) + fabsf((pd - pu) - (td - tu));
        }
        atomicAdd(&sL1, l1);
        atomicAdd(&sGR, gr);
    }

    // ---- SSIM via WMMA ----
    const int wave = threadIdx.x >> 5;
    const int lane = threadIdx.x & 31;
    const int hi   = lane >> 4;          // 0: K=0,1   1: K=2,3
    const int mn   = lane & 15;          // M (A-frag) or N (B-frag)
    const int kb   = hi * 2;
    const int wy   = (wave >> 2) * 16;   // wave tile row within block tile
    const int wx   = (wave & 3) * 16;    // wave tile col within block tile

    // gaussian band fragments: value g[j - mn]; serves as B for h-conv, A for v-conv
    v2f gfrag[7];
#pragma unroll
    for (int c = 0; c < 7; ++c) {
        const int j0 = 4 * c + kb;
        gfrag[c].x = gband(j0, mn);
        gfrag[c].y = gband(j0 + 1, mn);
    }

    v8f out[5];
#pragma unroll
    for (int ch = 0; ch < 5; ++ch) {
        // ---- horizontal conv: two 16-row WMMA tiles -> sTmp rows 0..31 ----
#pragma unroll
        for (int t2 = 0; t2 < 2; ++t2) {
            v8f acc = {};
            const int arow = wy + t2 * 16 + mn;   // A: both lane halves carry M=0..15
#pragma unroll
            for (int c = 0; c < 7; ++c) {
                const int col = wx + 4 * c + kb;
                const float p0 = sP[arow][col], p1 = sP[arow][col + 1];
                const float t0 = sT[arow][col], t1 = sT[arow][col + 1];
                float a0, a1;
                if      (ch == 0) { a0 = p0;      a1 = p1;      }
                else if (ch == 1) { a0 = t0;      a1 = t1;      }
                else if (ch == 2) { a0 = p0 * p0; a1 = p1 * p1; }
                else if (ch == 3) { a0 = t0 * t0; a1 = t1 * t1; }
                else              { a0 = p0 * t0; a1 = p1 * t1; }
                v2f a = { a0, a1 };
                acc = __builtin_amdgcn_wmma_f32_16x16x4_f32(
                        false, a, false, gfrag[c], (short)0, acc, false, false);
            }
            // C/D layout: VGPR v -> row v + 8*hi, col mn
#pragma unroll
            for (int v = 0; v < 8; ++v)
                sTmp[wave][t2 * 16 + v + hi * 8][mn] = acc[v];
        }
        // ---- vertical conv: out = G_band(16x28) x tmp(28x16) ----
        v8f vac = {};
#pragma unroll
        for (int c = 0; c < 7; ++c) {
            const int j0 = 4 * c + kb;
            v2f bfr;
            bfr.x = sTmp[wave][j0][mn];
            bfr.y = sTmp[wave][j0 + 1][mn];
            vac = __builtin_amdgcn_wmma_f32_16x16x4_f32(
                    false, gfrag[c], false, bfr, (short)0, vac, false, false);
        }
        out[ch] = vac;
    }

    // ---- per-pixel SSIM + reduction ----
    float lsum = 0.0f;
#pragma unroll
    for (int v = 0; v < 8; ++v) {
        const int oy = oy0 + wy + v + hi * 8;
        const int ox = ox0 + wx + mn;
        if (oy < OUT_H && ox < OUT_W) {
            const float mp  = out[0][v];
            const float mt  = out[1][v];
            const float vp  = out[2][v] - mp * mp;
            const float vt  = out[3][v] - mt * mt;
            const float cov = out[4][v] - mt * mp;
            const float lt  = (2.0f * mp * mt + C1_SSIM) / (mp * mp + mt * mt + C1_SSIM);
            const float sp  = sqrtf(vp);
            const float st  = sqrtf(vt);
            const float ct  = (2.0f * sp * st + C2_SSIM) / (vp + vt + C2_SSIM);
            const float s   = (cov + C3_SSIM) / (sp * st + C3_SSIM);
            lsum += lt * ct * s;
        }
    }
    atomicAdd(&sSS, lsum);
    __syncthreads();
    if (threadIdx.x == 0) {
        atomicAdd(&ws[0], sL1);
        atomicAdd(&ws[1], sGR);
        atomicAdd(&ws[2], sSS);
    }
}

// ---------------- kernel 2: combine ----------------
__global__ void dd_finalize_kernel(const float* __restrict__ ws, float* __restrict__ out) {
    if (threadIdx.x == 0 && blockIdx.x == 0) {
        const float l1   = ws[0] / 19660800.0f;           // 64*480*640
        const float gr   = ws[1] / 39321600.0f;           // 64*2*480*640
        const float ssim = ws[2] / 18950400.0f;           // 64*470*630
        out[0] = 0.1f * l1 + gr + 0.5f * (1.0f - ssim);
    }
}

extern "C" void kernel_launch(void* const* d_in, const int* in_sizes, int n_in,
                              void* d_out, int out_size, void* d_ws, size_t ws_size,
                              hipStream_t stream) {
    (void)in_sizes; (void)n_in; (void)out_size; (void)ws_size;
    const float* pred = (const float*)d_in[0];
    const float* targ = (const float*)d_in[1];
    float* out = (float*)d_out;
    float* ws  = (float*)d_ws;

    dd_init_kernel<<<1, 32, 0, stream>>>(ws);
    dd_fused_kernel<<<BATCH * TILES_X * TILES_Y, 256, 0, stream>>>(pred, targ, ws);
    dd_finalize_kernel<<<1, 32, 0, stream>>>(ws, out);
}